// LSTM_Op_910533067530
// MI455X (gfx1250) — compile-verified
//
#include <hip/hip_runtime.h>
#include <hip/hip_bf16.h>

typedef float v2f __attribute__((ext_vector_type(2)));
typedef float v8f __attribute__((ext_vector_type(8)));

#define NB  64
#define NIC 8
#define NID 256
#define NOC 8
#define NOD 256

static __device__ __forceinline__ v8f wmma_f32_16x16x4(v2f a, v2f b, v8f c) {
    // 8 args: (neg_a, A, neg_b, B, c_mod, C, reuse_a, reuse_b)
    return __builtin_amdgcn_wmma_f32_16x16x4_f32(false, a, false, b, (short)0, c, false, false);
}

// GEMM: Xs[g][b][o][u] = sum_{i,d} xOps[g,o,i,u,d]*x[b,i,d] + sum_{p,v} hOps[g,o,p,u,v]*h_p[b,p,v]
// One wave per (g, o, 32-row u-strip): 2 u-tiles x 4 b-tiles of 16x16 f32 accumulators.
__global__ __launch_bounds__(128, 1) void lstm_gemm(
    const float* __restrict__ x,     // [B, IC, ID]
    const float* __restrict__ h_p,   // [B, OC, OD]
    const float* __restrict__ xOps,  // [4, OC, IC, OD, ID]
    const float* __restrict__ hOps,  // [4, OC, OC, OD, OD]
    float* __restrict__ Xs)          // [4, B, OC, OD]
{
    const int lane  = threadIdx.x & 31;
    const int wid   = blockIdx.x * (blockDim.x >> 5) + (threadIdx.x >> 5); // 0..255
    const int upair = wid & 7;          // 8 strips of 32 rows cover OD=256
    const int oo    = (wid >> 3) & 7;   // out-channel
    const int gg    = wid >> 6;         // gate 0..3

    const int m  = lane & 15;           // M (or N) index within tile
    const int kk = (lane >> 4) << 1;    // K sub-offset: lanes 0-15 -> K0,K1 ; 16-31 -> K2,K3
    const int u0 = upair * 32;

    v8f acc[2][4];
    #pragma unroll
    for (int t = 0; t < 2; ++t)
        #pragma unroll
        for (int n = 0; n < 4; ++n)
            acc[t][n] = (v8f){0.f,0.f,0.f,0.f,0.f,0.f,0.f,0.f};

    const int go = gg * NOC + oo;
    const float* Wbase[2] = { xOps + (size_t)go * NIC * NOD * NID,
                              hOps + (size_t)go * NOC * NOD * NOD };
    const float* Abase[2] = { x, h_p };

    const int aoff = (u0 + m) * NID + kk;     // per-lane weight offset within (g,o,i) block
    const int boff = m * (NIC * NID) + kk;    // per-lane activation offset (batch-major)

    for (int phase = 0; phase < 2; ++phase) {
        const float* W = Wbase[phase];
        const float* A = Abase[phase];
        for (int i = 0; i < NIC; ++i) {
            const float* Wp0 = W + (size_t)i * NOD * NID + aoff;
            const float* Wp1 = Wp0 + 16 * NID;
            const float* Ap  = A + i * NID + boff;
            #pragma unroll 4
            for (int d = 0; d < NID; d += 4) {
                v2f a0 = *(const v2f*)(Wp0 + d);
                v2f a1 = *(const v2f*)(Wp1 + d);
                v2f b0 = *(const v2f*)(Ap + d);
                v2f b1 = *(const v2f*)(Ap + d + 16 * (NIC * NID));
                v2f b2 = *(const v2f*)(Ap + d + 32 * (NIC * NID));
                v2f b3 = *(const v2f*)(Ap + d + 48 * (NIC * NID));
                acc[0][0] = wmma_f32_16x16x4(a0, b0, acc[0][0]);
                acc[0][1] = wmma_f32_16x16x4(a0, b1, acc[0][1]);
                acc[0][2] = wmma_f32_16x16x4(a0, b2, acc[0][2]);
                acc[0][3] = wmma_f32_16x16x4(a0, b3, acc[0][3]);
                acc[1][0] = wmma_f32_16x16x4(a1, b0, acc[1][0]);
                acc[1][1] = wmma_f32_16x16x4(a1, b1, acc[1][1]);
                acc[1][2] = wmma_f32_16x16x4(a1, b2, acc[1][2]);
                acc[1][3] = wmma_f32_16x16x4(a1, b3, acc[1][3]);
            }
        }
    }

    // D layout (16x16 f32): VGPR r -> rows r (lanes 0-15) / r+8 (lanes 16-31), cols = lane&15.
    // The 8 per-lane elements are contiguous in u -> one aligned 32B store per tile.
    const int nlo = lane & 15;
    const int mhi = (lane >> 4) * 8;
    float* outG = Xs + (size_t)gg * NB * NOC * NOD;
    #pragma unroll
    for (int t = 0; t < 2; ++t) {
        #pragma unroll
        for (int bt = 0; bt < 4; ++bt) {
            const int b = bt * 16 + nlo;
            float* p = outG + ((size_t)b * NOC + oo) * NOD + (u0 + t * 16 + mhi);
            *(v8f*)p = acc[t][bt];
        }
    }
}

__device__ __forceinline__ float sigmoidf_(float v) {
    return 1.0f / (1.0f + __expf(-v));
}

__global__ __launch_bounds__(256) void lstm_gates(
    const float* __restrict__ Xs,   // [4, B*OC*OD]
    const float* __restrict__ c_p,  // [B, OC, OD]
    float* __restrict__ h,          // [B, OC, OD]
    int n)
{
    int j = blockIdx.x * blockDim.x + threadIdx.x;
    if (j >= n) return;
    float ig = sigmoidf_(Xs[j]);
    float fg = sigmoidf_(Xs[n + j]);
    float gt = tanhf(Xs[2 * n + j]);
    float og = sigmoidf_(Xs[3 * n + j]);
    float c  = fg * c_p[j] + ig * gt;
    h[j] = og * tanhf(c);
}

extern "C" void kernel_launch(void* const* d_in, const int* in_sizes, int n_in,
                              void* d_out, int out_size, void* d_ws, size_t ws_size,
                              hipStream_t stream) {
    const float* x    = (const float*)d_in[0];
    const float* c_p  = (const float*)d_in[1];
    const float* h_p  = (const float*)d_in[2];
    const float* xOps = (const float*)d_in[3];
    const float* hOps = (const float*)d_in[4];
    float* out = (float*)d_out;
    float* Xs  = (float*)d_ws;   // needs 4*64*8*256*4 B = 2 MB scratch

    lstm_gemm<<<64, 128, 0, stream>>>(x, h_p, xOps, hOps, Xs);
    const int n = NB * NOC * NOD; // 131072
    lstm_gates<<<(n + 255) / 256, 256, 0, stream>>>(Xs, c_p, out, n);
}